// Model_19069654794448
// MI455X (gfx1250) — compile-verified
//
#include <hip/hip_runtime.h>

// D = A(16x4) x B(4x16) + C(16x16), fp32 WMMA. Per-lane payloads:
//   A,B: 2 floats (64 elems / 32 lanes), C/D: 8 floats.
typedef __attribute__((ext_vector_type(2))) float v2f;
typedef __attribute__((ext_vector_type(8))) float v8f;

constexpr int kN = 512;   // state / matrix dim
constexpr int kT = 32;    // scan length

// One workgroup per batch element. 1024 threads = 32 waves; each wave owns one
// 16-column tile of the output vector. Memory-bound kernel: matrix elements are
// used exactly once -> non-temporal loads; carry vector double-buffered in LDS.
__global__ __launch_bounds__(1024)
void rnn_chain_wmma(const float* __restrict__ inp,     // [B, T, N, N]
                    const float* __restrict__ param,   // [N]
                    const float* __restrict__ offset,  // [N]
                    float* __restrict__ out)           // [B, N]
{
    __shared__ float vbuf[2][kN];   // ping-pong carry vector (4 KB)

    const int b    = blockIdx.x;
    const int tid  = threadIdx.x;
    const int lane = tid & 31;
    const int wave = tid >> 5;        // 0..31 -> one 16-col tile each
    const int half = lane >> 4;       // 0: lanes 0-15, 1: lanes 16-31
    const int n    = lane & 15;       // column within tile
    const int col  = (wave << 4) + n; // global output column
    const int k0   = half << 1;       // K base for this half-wave (0 or 2)

    // y0 = param + offset (same for every batch)
    for (int i = tid; i < kN; i += 1024)
        vbuf[0][i] = param[i] + offset[i];
    __syncthreads();

    const float* Mb = inp + (size_t)b * kT * kN * kN;

    float ylast = 0.0f;
    for (int t = 0; t < kT; ++t) {
        const float* __restrict__ M    = Mb + (size_t)t * kN * kN;
        const float* __restrict__ vcur = vbuf[t & 1];

        // y[col] = sum_i vcur[i] * M[i][col], accumulated 4 K-values per WMMA.
        // A: all 16 rows carry the same v-chunk  -> every D row equals y-tile.
        // ISA A layout: vgpr j, lanes<16 -> K=j; lanes>=16 -> K=2+j. B matches.
        v8f acc = {};
#pragma unroll 4
        for (int i = 0; i < kN; i += 4) {
            v2f a, bm;
            a[0] = vcur[i + k0];
            a[1] = vcur[i + k0 + 1];
            const float* p = M + (size_t)(i + k0) * kN + col;
            bm[0] = __builtin_nontemporal_load(p);        // row i+k0
            bm[1] = __builtin_nontemporal_load(p + kN);   // row i+k0+1
            acc = __builtin_amdgcn_wmma_f32_16x16x4_f32(
                /*neg_a=*/false, a, /*neg_b=*/false, bm,
                /*c_mod=*/(short)0, acc, /*reuse_a=*/false, /*reuse_b=*/false);
        }

        // D row 0 lives in acc[0] lanes 0-15 (and row 8, identical, lanes 16-31):
        // every lane's acc[0] == y[col].
        const float y = acc[0];
        ylast = y;

        // carry = leaky_relu(y, 0.1); write next buffer (16 writers per tile)
        if (t < kT - 1 && half == 0)
            vbuf[(t + 1) & 1][col] = (y > 0.0f) ? y : 0.1f * y;
        __syncthreads();   // writes visible + old buffer fully consumed
    }

    // reference returns the LAST pre-activation einsum result
    if (half == 0)
        out[(size_t)b * kN + col] = ylast;
}

extern "C" void kernel_launch(void* const* d_in, const int* in_sizes, int n_in,
                              void* d_out, int out_size, void* d_ws, size_t ws_size,
                              hipStream_t stream) {
    const float* inp    = (const float*)d_in[0];
    const float* param  = (const float*)d_in[1];
    const float* offset = (const float*)d_in[2];
    float* out          = (float*)d_out;

    const int B = in_sizes[0] / (kT * kN * kN);   // 32
    rnn_chain_wmma<<<B, 1024, 0, stream>>>(inp, param, offset, out);
}